// ChamferLikeDistanceLoss_31026843746910
// MI455X (gfx1250) — compile-verified
//
#include <hip/hip_runtime.h>
#include <math.h>

typedef __attribute__((ext_vector_type(2))) float v2f;
typedef __attribute__((ext_vector_type(8))) float v8f;

#define BATCHES 4
#define HW 64
#define NPTS 4096            // 64*64 per batch
#define TILES 256            // NPTS / 16
#define THREADS 256

// ---------------------------------------------------------------------------
// Pass 1: Sobel gradient magnitude, zero padding (matches lax conv pad=1).
// Kernel flip (conv vs corr) does not matter under sqrt(gx^2+gy^2).
// ---------------------------------------------------------------------------
__global__ __launch_bounds__(THREADS) void sobel_kernel(
    const float* __restrict__ depth, float* __restrict__ g) {
  int idx = blockIdx.x * THREADS + threadIdx.x;      // 0 .. 16383
  int w = idx & (HW - 1);
  int h = (idx >> 6) & (HW - 1);
  int b = idx >> 12;
  const float* p = depth + b * NPTS;

  auto at = [&](int hh, int ww) -> float {
    return (hh < 0 || hh >= HW || ww < 0 || ww >= HW) ? 0.0f : p[hh * HW + ww];
  };
  float tl = at(h - 1, w - 1), tm = at(h - 1, w), tr = at(h - 1, w + 1);
  float ml = at(h, w - 1),                         mr = at(h, w + 1);
  float bl = at(h + 1, w - 1), bm = at(h + 1, w), br = at(h + 1, w + 1);

  float gx = (tl - tr) + 2.0f * (ml - mr) + (bl - br);
  float gy = (tl + 2.0f * tm + tr) - (bl + 2.0f * bm + br);
  g[idx] = sqrtf(gx * gx + gy * gy + 1e-8f);
}

// ---------------------------------------------------------------------------
// Pass 2: pairwise |x_i - y_j| tile mins via V_WMMA_F32_16X16X4_F32.
//   D = A(16x4) x B(4x16),  A rows = [x_i, -1, 0, 0], B rows = [1; y; 0; 0]
//   => D[i][j] = x_i - y_j  (exact in f32)
// grid: (rowblock 0..31, batch 0..3, direction 0..1); 8 waves/block, one
// 16-row A-tile per wave, looping all 256 column tiles with Y staged in LDS.
// ---------------------------------------------------------------------------
__global__ __launch_bounds__(THREADS) void chamfer_tile_kernel(
    const float* __restrict__ g, const float* __restrict__ bgt,
    float* __restrict__ partial) {
  __shared__ float sY[NPTS];   // 16 KB

  const int rowblk = blockIdx.x;   // 0..31
  const int batch  = blockIdx.y;   // 0..3
  const int dir    = blockIdx.z;   // 0: rows = g (dist1), 1: rows = bgt (dist2)

  const float* Xb = (dir == 0 ? g : bgt) + batch * NPTS;
  const float* Yb = (dir == 0 ? bgt : g) + batch * NPTS;

  for (int k = threadIdx.x; k < NPTS; k += THREADS) sY[k] = Yb[k];
  __syncthreads();

  const int wave = threadIdx.x >> 5;
  const int lane = threadIdx.x & 31;
  const int tile = rowblk * 8 + wave;    // A-tile index 0..255
  const int row0 = tile * 16;
  const bool lo  = lane < 16;            // lanes 0-15 carry K=0,1; 16-31 carry K=2,3 (zero)

  // Loop-invariant A fragment: lane M holds {x_M, -1} (low half), {0,0} (high half)
  v2f a;
  a.x = lo ? Xb[row0 + lane] : 0.0f;
  a.y = lo ? -1.0f : 0.0f;

  v8f m;
#pragma unroll
  for (int r = 0; r < 8; ++r) m[r] = 3.4e38f;

  const int laneN = lane & 15;
  for (int t = 0; t < TILES; ++t) {
    float yv = sY[t * 16 + laneN];
    v2f bb;
    bb.x = lo ? 1.0f : 0.0f;   // B row K=0 = ones (K=2 = 0 on high lanes)
    bb.y = lo ? yv : 0.0f;     // B row K=1 = y    (K=3 = 0 on high lanes)
    v8f c = {};
    c = __builtin_amdgcn_wmma_f32_16x16x4_f32(
        /*neg_a=*/false, a, /*neg_b=*/false, bb,
        /*c_mod=*/(short)0, c, /*reuse_a=*/false, /*reuse_b=*/false);
#pragma unroll
    for (int r = 0; r < 8; ++r) m[r] = fminf(m[r], fabsf(c[r]));
  }

  // Row-min: reduce across each 16-lane half (masks 1,2,4,8 stay in-half).
#pragma unroll
  for (int r = 0; r < 8; ++r) {
    float v = m[r];
    v = fminf(v, __shfl_xor(v, 1, 32));
    v = fminf(v, __shfl_xor(v, 2, 32));
    v = fminf(v, __shfl_xor(v, 4, 32));
    v = fminf(v, __shfl_xor(v, 8, 32));
    m[r] = v;
  }
  float s = 0.0f;
#pragma unroll
  for (int r = 0; r < 8; ++r) s += m[r];   // lane0: rows 0-7, lane16: rows 8-15
  float tot = __shfl(s, 0, 32) + __shfl(s, 16, 32);
  if (lane == 0) partial[(dir * BATCHES + batch) * TILES + tile] = tot;
}

// ---------------------------------------------------------------------------
// Pass 3: deterministic fixed-order reduction of 2048 partials; mean scale.
// ---------------------------------------------------------------------------
__global__ __launch_bounds__(THREADS) void reduce_kernel(
    const float* __restrict__ partial, float* __restrict__ out) {
  __shared__ float sm[THREADS];
  float s = 0.0f;
#pragma unroll
  for (int k = 0; k < 8; ++k) s += partial[threadIdx.x + k * THREADS];
  sm[threadIdx.x] = s;
  __syncthreads();
  for (int off = THREADS / 2; off > 0; off >>= 1) {
    if (threadIdx.x < off) sm[threadIdx.x] += sm[threadIdx.x + off];
    __syncthreads();
  }
  if (threadIdx.x == 0) out[0] = sm[0] / (float)(BATCHES * NPTS);
}

extern "C" void kernel_launch(void* const* d_in, const int* in_sizes, int n_in,
                              void* d_out, int out_size, void* d_ws, size_t ws_size,
                              hipStream_t stream) {
  const float* depth = (const float*)d_in[0];   // (4,1,64,64) f32
  const float* bgt   = (const float*)d_in[1];   // (4,1,64,64) f32
  float* g       = (float*)d_ws;                // 16384 floats (64 KB)
  float* partial = g + BATCHES * NPTS;          // 2048 floats
  float* out     = (float*)d_out;

  sobel_kernel<<<BATCHES * NPTS / THREADS, THREADS, 0, stream>>>(depth, g);

  dim3 grid(32, BATCHES, 2);
  chamfer_tile_kernel<<<grid, THREADS, 0, stream>>>(g, bgt, partial);

  reduce_kernel<<<1, THREADS, 0, stream>>>(partial, out);
}